// dygraphSAGE_46772193853696
// MI455X (gfx1250) — compile-verified
//
#include <hip/hip_runtime.h>
#include <hip/hip_bf16.h>

typedef __attribute__((ext_vector_type(2))) float v2f;
typedef __attribute__((ext_vector_type(8))) float v8f;

#define FD 128          // feature dim
#define LRELU 0.2f

static __device__ __forceinline__ v8f wmma4(v2f a, v2f b, v8f c) {
    // D = A(16x4) * B(4x16) + C, f32 in / f32 acc
    return __builtin_amdgcn_wmma_f32_16x16x4_f32(false, a, false, b, (short)0, c, false, false);
}

// ---------------- degree ----------------
__global__ void deg_kernel(const int* __restrict__ ei, float* __restrict__ deg, int E) {
    int e = blockIdx.x * blockDim.x + threadIdx.x;
    if (e < E) atomicAdd(&deg[ei[E + e]], 1.0f);
}

__global__ void invdeg_kernel(float* __restrict__ deg, int N) {
    int i = blockIdx.x * blockDim.x + threadIdx.x;
    if (i < N) deg[i] = 1.0f / fmaxf(deg[i], 1.0f);
}

// ---------------- edge scatter: agg[dst] += h[src] ----------------
// 32 lanes per edge, float4 per lane (128 cols)
__global__ void scatter_kernel(const float* __restrict__ h, const int* __restrict__ ei,
                               float* __restrict__ agg, int E) {
    long long t = (long long)blockIdx.x * blockDim.x + threadIdx.x;
    long long e = t >> 5;
    if (e >= E) return;
    int q = (int)(t & 31) * 4;
    size_t s = (size_t)ei[e];
    size_t d = (size_t)ei[E + e];
    const float4 v = *(const float4*)(h + s * FD + q);
    float* a = agg + d * FD + q;
    atomicAdd(a + 0, v.x);
    atomicAdd(a + 1, v.y);
    atomicAdd(a + 2, v.z);
    atomicAdd(a + 3, v.w);
}

// ---------------- SAGE layer: out = relu([h | agg*invdeg] @ W), W:[256,128] ----------------
// block = 256 thr = 8 waves; block tile = 16 rows x 128 cols; wave tile = 16x16
__global__ __launch_bounds__(256)
void sage_gemm_kernel(const float* __restrict__ H, const float* __restrict__ AGG,
                      const float* __restrict__ invdeg, const float* __restrict__ W,
                      float* __restrict__ out, int nrows) {
    __shared__ float ldsA[16 * 256];
    const int tid = threadIdx.x;
    const int row0 = blockIdx.x * 16;

    // stage A = [h | agg/deg] 16x256 into LDS
    for (int i = tid; i < 16 * 256; i += 256) {
        int m = i >> 8, k = i & 255;
        int row = row0 + m;
        float v = 0.0f;
        if (row < nrows) {
            if (k < FD) v = H[(size_t)row * FD + k];
            else        v = AGG[(size_t)row * FD + (k - FD)] * invdeg[row];
        }
        ldsA[i] = v;
    }
    __syncthreads();

    const int lane = tid & 31, lo = lane & 15, hi = lane >> 4;
    const int n0 = (tid >> 5) * 16;          // wave's 16-column slab
    const int col = n0 + lo;
    const float* Ap = &ldsA[lo * 256 + 2 * hi];

    v8f c = {};
    #pragma unroll 8
    for (int ks = 0; ks < 64; ++ks) {
        int k = 4 * ks + 2 * hi;
        v2f a = *(const v2f*)(Ap + 4 * ks);
        float b0 = W[(size_t)k * FD + col];
        float b1 = W[(size_t)(k + 1) * FD + col];
        v2f b = {b0, b1};
        c = wmma4(a, b, c);
    }

    #pragma unroll
    for (int r = 0; r < 8; ++r) {
        int row = row0 + r + 8 * hi;
        if (row < nrows) {
            float v = c[r];
            out[(size_t)row * FD + col] = v > 0.0f ? v : 0.0f;
        }
    }
}

// ---------------- temporal fusion over first `num` rows (in place on `out`) ----------------
// tf   = ([feat | past0 | past1] @ [W0; Wp0; Wp1]) * (1/G)         (K = 384)
// f    = leakyrelu([feat | tf] @ W_T) ; f /= max(||f||, 1e-12)     (K = 256)
__global__ __launch_bounds__(256)
void fuse_kernel(float* __restrict__ out, const float* __restrict__ past,
                 const float* __restrict__ W0, const float* __restrict__ Wp,
                 const float* __restrict__ WT, int num, float inv_g) {
    __shared__ float ldsA[16 * 384];
    __shared__ float ldsT[16 * 128];
    __shared__ float ldsInv[16];
    const int tid = threadIdx.x;
    const int row0 = blockIdx.x * 16;

    // stage A = [feat | past0 | past1] 16x384
    for (int i = tid; i < 16 * 384; i += 256) {
        int m = i / 384, k = i % 384;
        int row = row0 + m;
        float v = 0.0f;
        if (row < num) {
            if (k < 128)      v = out[(size_t)row * FD + k];
            else if (k < 256) v = past[(size_t)row * FD + (k - 128)];
            else              v = past[((size_t)num + row) * FD + (k - 256)];
        }
        ldsA[i] = v;
    }
    __syncthreads();

    const int lane = tid & 31, lo = lane & 15, hi = lane >> 4;
    const int n0 = (tid >> 5) * 16;
    const int col = n0 + lo;

    // ---- stage 1: time_feat (K = 384), B = [W0 ; Wp(flat 256x128)] ----
    v8f c = {};
    #pragma unroll 4
    for (int ks = 0; ks < 96; ++ks) {
        int k = 4 * ks + 2 * hi;
        v2f a = *(const v2f*)&ldsA[lo * 384 + k];
        const float* Bp = (k < 128) ? (W0 + (size_t)k * FD)
                                    : (Wp + (size_t)(k - 128) * FD);
        v2f b = {Bp[col], Bp[FD + col]};
        c = wmma4(a, b, c);
    }
    #pragma unroll
    for (int r = 0; r < 8; ++r) {
        int row = r + 8 * hi;
        ldsT[row * 128 + col] = c[r] * inv_g;
    }
    __syncthreads();

    // ---- stage 2: fused (K = 256), A = [feat | tf], B = W_T ----
    v8f d = {};
    #pragma unroll 4
    for (int ks = 0; ks < 64; ++ks) {
        int k = 4 * ks + 2 * hi;
        v2f a = (k < 128) ? *(const v2f*)&ldsA[lo * 384 + k]
                          : *(const v2f*)&ldsT[lo * 128 + (k - 128)];
        v2f b = {WT[(size_t)k * FD + col], WT[(size_t)(k + 1) * FD + col]};
        d = wmma4(a, b, d);
    }
    __syncthreads();   // everyone done reading ldsT

    // leaky relu -> ldsT
    #pragma unroll
    for (int r = 0; r < 8; ++r) {
        int row = r + 8 * hi;
        float v = d[r];
        ldsT[row * 128 + col] = v > 0.0f ? v : LRELU * v;
    }
    __syncthreads();

    // per-row L2 norm
    if (tid < 16) {
        float s = 0.0f;
        #pragma unroll 4
        for (int j = 0; j < 128; ++j) {
            float v = ldsT[tid * 128 + j];
            s += v * v;
        }
        ldsInv[tid] = 1.0f / fmaxf(sqrtf(s), 1e-12f);
    }
    __syncthreads();

    for (int i = tid; i < 16 * 128; i += 256) {
        int m = i >> 7, j = i & 127;
        int row = row0 + m;
        if (row < num) out[(size_t)row * FD + j] = ldsT[i] * ldsInv[m];
    }
}

extern "C" void kernel_launch(void* const* d_in, const int* in_sizes, int n_in,
                              void* d_out, int out_size, void* d_ws, size_t ws_size,
                              hipStream_t stream) {
    const float* x    = (const float*)d_in[0];
    const float* past = (const float*)d_in[1];
    const float* W1   = (const float*)d_in[2];
    const float* W2   = (const float*)d_in[3];
    const float* W0   = (const float*)d_in[4];
    const float* Wp   = (const float*)d_in[5];
    const float* WT   = (const float*)d_in[6];
    const int*   ei   = (const int*)d_in[7];

    const int N   = in_sizes[0] / FD;
    const int E   = in_sizes[7] / 2;
    const int Gm1 = in_sizes[5] / (FD * FD);          // 2
    const int num = in_sizes[1] / (Gm1 * FD);         // 50000
    const float inv_g = 1.0f / (float)(Gm1 + 1);

    float* out    = (float*)d_out;
    float* invdeg = (float*)d_ws;
    size_t npad   = ((size_t)N + 255) & ~(size_t)255;
    float* agg    = invdeg + npad;
    float* feat1  = agg + (size_t)N * FD;

    hipMemsetAsync(invdeg, 0, (size_t)N * sizeof(float), stream);
    hipMemsetAsync(agg, 0, (size_t)N * FD * sizeof(float), stream);

    deg_kernel<<<(E + 255) / 256, 256, 0, stream>>>(ei, invdeg, E);
    invdeg_kernel<<<(N + 255) / 256, 256, 0, stream>>>(invdeg, N);

    unsigned sblocks = (unsigned)(((long long)E * 32 + 255) / 256);
    unsigned gblocks = (unsigned)((N + 15) / 16);
    unsigned fblocks = (unsigned)((num + 15) / 16);

    // layer 1
    scatter_kernel<<<sblocks, 256, 0, stream>>>(x, ei, agg, E);
    sage_gemm_kernel<<<gblocks, 256, 0, stream>>>(x, agg, invdeg, W1, feat1, N);

    // layer 2 (writes feat2 directly into d_out)
    hipMemsetAsync(agg, 0, (size_t)N * FD * sizeof(float), stream);
    scatter_kernel<<<sblocks, 256, 0, stream>>>(feat1, ei, agg, E);
    sage_gemm_kernel<<<gblocks, 256, 0, stream>>>(feat1, agg, invdeg, W2, out, N);

    // temporal fusion + normalize over first `num` rows, in place
    fuse_kernel<<<fblocks, 256, 0, stream>>>(out, past, W0, Wp, WT, num, inv_g);
}